// PSOGNN_5119601017232
// MI455X (gfx1250) — compile-verified
//
#include <hip/hip_runtime.h>
#include <math.h>

// GCN dims (fixed by the reference problem)
#define DHID 32     // hidden width H
#define DIN  128    // input feature dim
#define NOUT 3

typedef float v2f __attribute__((ext_vector_type(2)));
typedef float v8f __attribute__((ext_vector_type(8)));

// ---------------------------------------------------------------- utilities
__global__ void zero_f32(float* __restrict__ p, int n) {
    int i = blockIdx.x * blockDim.x + threadIdx.x;
    if (i < n) p[i] = 0.0f;
}

// degree counted at dst (PyG convention), +1 self loop applied in make_norm
__global__ void deg_count(const int* __restrict__ dst, int E, float* __restrict__ deg) {
    int e = blockIdx.x * blockDim.x + threadIdx.x;
    if (e < E) atomicAdd(&deg[dst[e]], 1.0f);
}

__global__ void make_norm(float* __restrict__ deg_norm, int n) {
    int i = blockIdx.x * blockDim.x + threadIdx.x;
    if (i < n) deg_norm[i] = rsqrtf(deg_norm[i] + 1.0f);
}

// ---------------------------------------------------------------- WMMA GEMM
// out[row, 0:32] = norm[row] * (A[row, 0:K] @ W[0:K, 0:32])
// One wave32 per 16-row tile. Two 16x16 f32 accumulators cover all 32 cols.
// f32 WMMA 16x16x4: A lane layout -> lanes 0-15: M=lane, K={k,k+1};
//                                    lanes 16-31: M=lane-16, K={k+2,k+3}.
// C/D layout: VGPR r -> row M = r + (lane/16)*8, col N = lane%16.
// Rows are CLAMPED (not exec-predicated) so EXEC stays all-ones for WMMA and
// the A-stream is a clean global_load_b64 per k-step.
template <int K>
__global__ __launch_bounds__(32) void gemm_norm(const float* __restrict__ A,
                                                const float* __restrict__ W,
                                                const float* __restrict__ norm,
                                                float* __restrict__ out,
                                                int nrows) {
    const int lane = threadIdx.x & 31;
    const int half = lane >> 4;     // 0 or 1
    const int m    = lane & 15;
    const int base = blockIdx.x * 16;
    int row = base + m;
    if (row >= nrows) row = nrows - 1;   // clamp: keep loads in-bounds, EXEC uniform

    const float* __restrict__ arow = A + (size_t)row * K + half * 2;

    v8f acc0 = {};
    v8f acc1 = {};

    #pragma unroll
    for (int k = 0; k < K; k += 4) {
        // packed A fragment: 8B-aligned (k%4==0, half*2 even, row stride mult of 8)
        const v2f a = *(const v2f*)(arow + k);

        const int ka = k + half * 2;
        // B fragment (K x 16 tile of W): col N = m (+16), K = ka, ka+1
        v2f b0, b1;
        b0.x = W[(size_t)(ka + 0) * DHID + m];
        b0.y = W[(size_t)(ka + 1) * DHID + m];
        b1.x = W[(size_t)(ka + 0) * DHID + 16 + m];
        b1.y = W[(size_t)(ka + 1) * DHID + 16 + m];

        acc0 = __builtin_amdgcn_wmma_f32_16x16x4_f32(false, a, false, b0,
                                                     (short)0, acc0, false, false);
        acc1 = __builtin_amdgcn_wmma_f32_16x16x4_f32(false, a, false, b1,
                                                     (short)0, acc1, false, false);
    }

    #pragma unroll
    for (int r = 0; r < 8; ++r) {
        const int orow = base + half * 8 + r;
        if (orow < nrows) {
            const float nr = norm[orow];
            out[(size_t)orow * DHID + m]      = acc0[r] * nr;
            out[(size_t)orow * DHID + 16 + m] = acc1[r] * nr;
        }
    }
}

// ---------------------------------------------------------------- scatter
// agg[dst, c] += hn[src, c]  -- one edge per wave32, lane = channel.
__global__ void scatter_edges(const int* __restrict__ src, const int* __restrict__ dst,
                              const float* __restrict__ hn, float* __restrict__ agg,
                              int E) {
    const int wave = blockIdx.x * (blockDim.x >> 5) + (threadIdx.x >> 5);
    const int lane = threadIdx.x & 31;
    if (wave < E) {
        const int s = src[wave];
        const int d = dst[wave];
        const float v = hn[(size_t)s * DHID + lane];
        atomicAdd(&agg[(size_t)d * DHID + lane], v);
    }
}

// ---------------------------------------------------------------- post-pass
// act = relu(norm[i] * (agg + hn) + bias[c])
__global__ void post_relu(const float* __restrict__ hn, const float* __restrict__ agg,
                          const float* __restrict__ norm, const float* __restrict__ bias,
                          float* __restrict__ act, int n) {
    const int idx = blockIdx.x * blockDim.x + threadIdx.x;
    if (idx < n * DHID) {
        const int i = idx >> 5;       // node  (DHID == 32)
        const int c = idx & 31;       // channel
        float v = norm[i] * (agg[idx] + hn[idx]) + bias[c];
        act[idx] = v > 0.0f ? v : 0.0f;
    }
}

// ---------------------------------------------------------------- final FC
__global__ void final_fc(const float* __restrict__ act, const float* __restrict__ Wfc,
                         const float* __restrict__ bfc, float* __restrict__ out, int n) {
    const int i = blockIdx.x * blockDim.x + threadIdx.x;
    if (i < n) {
        float a0 = bfc[0], a1 = bfc[1], a2 = bfc[2];
        #pragma unroll
        for (int k = 0; k < DHID; ++k) {
            const float v = act[(size_t)i * DHID + k];
            a0 += v * Wfc[k * NOUT + 0];
            a1 += v * Wfc[k * NOUT + 1];
            a2 += v * Wfc[k * NOUT + 2];
        }
        out[(size_t)i * NOUT + 0] = 1.0f / (1.0f + expf(-a0));
        out[(size_t)i * NOUT + 1] = 1.0f / (1.0f + expf(-a1));
        out[(size_t)i * NOUT + 2] = 1.0f / (1.0f + expf(-a2));
    }
}

// ---------------------------------------------------------------- launcher
extern "C" void kernel_launch(void* const* d_in, const int* in_sizes, int n_in,
                              void* d_out, int out_size, void* d_ws, size_t ws_size,
                              hipStream_t stream) {
    const float* x   = (const float*)d_in[0];
    const int*   ei  = (const int*)  d_in[1];
    const float* W1  = (const float*)d_in[2];
    const float* b1  = (const float*)d_in[3];
    const float* W2  = (const float*)d_in[4];
    const float* b2  = (const float*)d_in[5];
    const float* Wfc = (const float*)d_in[6];
    const float* bfc = (const float*)d_in[7];

    const int N = in_sizes[0] / DIN;
    const int E = in_sizes[1] / 2;
    const int* src = ei;
    const int* dst = ei + E;

    float* ws   = (float*)d_ws;
    float* norm = ws;                        // N       (deg, then rsqrt in place)
    float* hn   = norm + N;                  // N*32    (norm-scaled transformed feats)
    float* agg  = hn + (size_t)N * DHID;     // N*32    (scatter accumulator)
    float* act  = agg + (size_t)N * DHID;    // N*32    (layer activation)

    const int TB = 256;
    const int nh = N * DHID;
    const int gemmBlocks   = (N + 15) / 16;
    const int wavesPerBlk  = TB / 32;
    const int scatterBlks  = (E + wavesPerBlk - 1) / wavesPerBlk;

    // degree + norm
    zero_f32<<<(N + TB - 1) / TB, TB, 0, stream>>>(norm, N);
    zero_f32<<<(nh + TB - 1) / TB, TB, 0, stream>>>(agg, nh);
    deg_count<<<(E + TB - 1) / TB, TB, 0, stream>>>(dst, E, norm);
    make_norm<<<(N + TB - 1) / TB, TB, 0, stream>>>(norm, N);

    // ---- layer 1
    gemm_norm<DIN><<<gemmBlocks, 32, 0, stream>>>(x, W1, norm, hn, N);
    scatter_edges<<<scatterBlks, TB, 0, stream>>>(src, dst, hn, agg, E);
    post_relu<<<(nh + TB - 1) / TB, TB, 0, stream>>>(hn, agg, norm, b1, act, N);

    // ---- layer 2
    gemm_norm<DHID><<<gemmBlocks, 32, 0, stream>>>(act, W2, norm, hn, N);
    zero_f32<<<(nh + TB - 1) / TB, TB, 0, stream>>>(agg, nh);
    scatter_edges<<<scatterBlks, TB, 0, stream>>>(src, dst, hn, agg, E);
    post_relu<<<(nh + TB - 1) / TB, TB, 0, stream>>>(hn, agg, norm, b2, act, N);

    // ---- readout
    final_fc<<<(N + TB - 1) / TB, TB, 0, stream>>>(act, Wfc, bfc, (float*)d_out, N);
}